// QuantumConv2D_10325101380309
// MI455X (gfx1250) — compile-verified
//
#include <hip/hip_runtime.h>

// ---------------------------------------------------------------------------
// QuantumConv2D — closed form.
//
// Heisenberg-picture reduction of the reference circuit:
//   out[b,c,i,j] = cos(theta[9]) * prod_{di,dj in 0..2} cos(x[b,c,i+di,j+dj])
//
// Memory-bound (~1.5 MB total traffic -> ~65 ns at 23.3 TB/s). CDNA5 path:
// async global->LDS DMA for the halo tile (ASYNCcnt), DS product pass.
// WMMA intentionally NOT used: a 3x3 *product* stencil is not a matmul.
// ---------------------------------------------------------------------------

#ifndef __has_builtin
#define __has_builtin(x) 0
#endif

#if __has_builtin(__builtin_amdgcn_global_load_async_to_lds_b32)
#define QC_ASYNC_LDS 1
#else
#define QC_ASYNC_LDS 0
#endif

namespace {
constexpr int Himg = 64;
constexpr int Wimg = 64;
constexpr int Ho   = 62;   // (64 - 3)/1 + 1
constexpr int Wo   = 62;
constexpr int TILE = 16;   // output tile edge
constexpr int HALO = 18;   // TILE + 2
constexpr int LDSW = 20;   // padded LDS row stride (dwords) to avoid bank conflicts
constexpr int PLANES = 16 * 3;        // B * C
constexpr int TILES_PER_PLANE = 16;   // 4 x 4 tiles cover 62x62
}  // namespace

// Builtin signature (from hipcc diagnostic): arg0 is addrspace(1) int*
// (printed as "__device__ int *"); LDS side is addrspace(3) int*.
typedef __attribute__((address_space(1))) int __as1_int;
typedef __attribute__((address_space(3))) int __as3_int;

__global__ __launch_bounds__(256)
void qconv3x3_cosprod(const float* __restrict__ x,
                      const float* __restrict__ theta,
                      float* __restrict__ out) {
  __shared__ float rawA[HALO * LDSW];  // staged raw pixels (halo tile)
  __shared__ float cosB[HALO * LDSW];  // cos(pixel)

  const int t     = threadIdx.x;
  const int blk   = blockIdx.x;
  const int plane = blk >> 4;          // b*C + c, 0..47
  const int tile  = blk & 15;
  const int row0  = (tile >> 2) * TILE;
  const int col0  = (tile & 3)  * TILE;

  const float* src = x + plane * (Himg * Wimg);

  // ---- Stage A: global -> LDS halo tile (18x18, clamped at image edge) ----
  for (int i = t; i < HALO * HALO; i += 256) {
    const int r  = i / HALO;
    const int c  = i - r * HALO;
    const int gr = row0 + r;
    const int gc = col0 + c;
    if (gr < Himg && gc < Wimg) {
#if QC_ASYNC_LDS
      // CDNA5 async copy: per-lane b32, tracked by ASYNCcnt, no VGPR data.
      __builtin_amdgcn_global_load_async_to_lds_b32(
          (__as1_int*)(src + gr * Wimg + gc),
          (__as3_int*)(&rawA[r * LDSW + c]),
          /*imm offset=*/0, /*cpol=*/0);
#else
      rawA[r * LDSW + c] = src[gr * Wimg + gc];
#endif
    }
  }
#if QC_ASYNC_LDS
#if __has_builtin(__builtin_amdgcn_s_wait_asynccnt)
  __builtin_amdgcn_s_wait_asynccnt(0);
#else
  asm volatile("s_wait_asynccnt 0" ::: "memory");
#endif
#endif
  __syncthreads();

  // ---- Stage B: cos once per staged pixel (324/tile, not 9 per output) ----
  for (int i = t; i < HALO * HALO; i += 256) {
    const int r = i / HALO;
    const int c = i - r * HALO;
    const int k = r * LDSW + c;
    cosB[k] = __builtin_cosf(rawA[k]);
  }
  __syncthreads();

  // ---- Stage C: one output per thread, 3x3 product from LDS ----
  const int r  = t >> 4;
  const int c  = t & 15;
  const int oy = row0 + r;
  const int ox = col0 + c;
  if (oy < Ho && ox < Wo) {
    const float* b0 = &cosB[r * LDSW + c];
    float p = b0[0]            * b0[1]            * b0[2]
            * b0[LDSW + 0]     * b0[LDSW + 1]     * b0[LDSW + 2]
            * b0[2 * LDSW + 0] * b0[2 * LDSW + 1] * b0[2 * LDSW + 2];
    // Only theta[9] (the ancilla's RY angle) survives the Heisenberg algebra.
    out[plane * (Ho * Wo) + oy * Wo + ox] = p * __builtin_cosf(theta[9]);
  }
}

extern "C" void kernel_launch(void* const* d_in, const int* in_sizes, int n_in,
                              void* d_out, int out_size, void* d_ws, size_t ws_size,
                              hipStream_t stream) {
  (void)in_sizes; (void)n_in; (void)d_ws; (void)ws_size; (void)out_size;
  const float* x     = (const float*)d_in[0];   // [16,3,64,64] f32
  const float* theta = (const float*)d_in[1];   // [10] f32
  float*       out   = (float*)d_out;           // [16,3,62,62] f32

  qconv3x3_cosprod<<<dim3(PLANES * TILES_PER_PLANE), dim3(256), 0, stream>>>(
      x, theta, out);
}